// Attention_28321014349827
// MI455X (gfx1250) — compile-verified
//
#include <hip/hip_runtime.h>
#include <hip/hip_bf16.h>

typedef __attribute__((ext_vector_type(2))) float v2f;
typedef __attribute__((ext_vector_type(8))) float v8f;

static constexpr int kB = 4;
static constexpr int kN = 2048;
static constexpr int kD = 512;       // model dim
static constexpr int kH = 8;
static constexpr int kDh = 64;
static constexpr int kT = 1536;      // 3 * inner

__device__ __forceinline__ v8f wmma4(v2f a, v2f b, v8f c) {
  // V_WMMA_F32_16X16X4_F32 : D = A(16x4,f32) * B(4x16,f32) + C(16x16,f32)
  return __builtin_amdgcn_wmma_f32_16x16x4_f32(false, a, false, b, (short)0, c,
                                               false, false);
}

// Async global->LDS 16B per lane (GLOBAL_LOAD_ASYNC_TO_LDS_B128, ASYNCcnt).
__device__ __forceinline__ void async_cp16(void* lds_dst, const void* gsrc) {
  uint32_t lo = (uint32_t)(uintptr_t)lds_dst;  // LDS offset = low 32 bits
  asm volatile("global_load_async_to_lds_b128 %0, %1, off"
               :: "v"(lo), "v"(gsrc)
               : "memory");
}
__device__ __forceinline__ void wait_async0() {
  asm volatile("s_wait_asynccnt 0" ::: "memory");
}

// ---------------------------------------------------------------------------
// C[M,N] = A[M,K] @ B[K,N] (+ R) — 256 thr = 8 waves, tile 64x32, BK=32,
// async-LDS double-buffered staging.
// ---------------------------------------------------------------------------
#define LDA 36  // pad: b128-store aligned (144B rows), banks 36l%64 distinct
__global__ __launch_bounds__(256) void gemm_f32_wmma(
    const float* __restrict__ A, const float* __restrict__ B,
    const float* __restrict__ R, float* __restrict__ C, int M, int N, int K) {
  __shared__ float As[2][64 * LDA];
  __shared__ float Bs[2][32 * LDA];
  const int t = threadIdx.x;
  const int lane = t & 31;
  const int w = t >> 5;
  const int lcol = lane & 15;
  const int koff = (lane < 16) ? 0 : 2;   // WMMA A/B: lane half selects K-half
  const int m0 = blockIdx.y * 64;
  const int n0 = blockIdx.x * 32;
  const int mW = (w >> 1) * 16;
  const int nW = (w & 1) * 16;

  // staging coords (float4 granularity)
  const int ra0 = t >> 3, ca = (t & 7) * 4;          // A: 2 chunks of 64x32
  const int rb = t >> 3, cb = (t & 7) * 4;           // B: 32x32, 1 per thread

  auto stage = [&](int buf, int kt) {
    async_cp16(&As[buf][ra0 * LDA + ca], &A[(size_t)(m0 + ra0) * K + kt + ca]);
    async_cp16(&As[buf][(ra0 + 32) * LDA + ca],
               &A[(size_t)(m0 + ra0 + 32) * K + kt + ca]);
    async_cp16(&Bs[buf][rb * LDA + cb], &B[(size_t)(kt + rb) * N + n0 + cb]);
  };

  v8f acc = {};
  stage(0, 0);
  int buf = 0;
  for (int kt = 0; kt < K; kt += 32) {
    wait_async0();
    __syncthreads();
    if (kt + 32 < K) stage(buf ^ 1, kt + 32);
    // preload 8 fragment sets, then 8 back-to-back WMMAs
    const v2f* Arow = (const v2f*)&As[buf][(mW + lcol) * LDA];
    const float* Bcol = &Bs[buf][nW + lcol];
    v2f af[8];
    float bx[8], by[8];
#pragma unroll
    for (int kk = 0; kk < 8; ++kk) {
      int kidx = kk * 4 + koff;
      af[kk] = Arow[kidx >> 1];
      bx[kk] = Bcol[kidx * LDA];
      by[kk] = Bcol[(kidx + 1) * LDA];
    }
#pragma unroll
    for (int kk = 0; kk < 8; ++kk) {
      v2f b;
      b.x = bx[kk];
      b.y = by[kk];
      acc = wmma4(af[kk], b, acc);
    }
    __syncthreads();
    buf ^= 1;
  }
  const int rowOff = (lane < 16) ? 0 : 8;  // C layout: lanes16-31 -> rows v+8
#pragma unroll
  for (int v = 0; v < 8; ++v) {
    size_t idx = (size_t)(m0 + mW + v + rowOff) * N + n0 + nW + lcol;
    float val = acc[v];
    if (R) val += R[idx];
    C[idx] = val;
  }
}

// ---------------------------------------------------------------------------
// Per (b,h): S = scale * K^T V (64x64) and csum = colsum(V) (64)
// Async-LDS staged 16-key tiles, double buffered.
// ---------------------------------------------------------------------------
#define LDK 68  // 272B rows: b128 aligned; fragment banks 4l%64 distinct
__global__ __launch_bounds__(256) void kv_stats(const float* __restrict__ qkv,
                                                float* __restrict__ S,
                                                float* __restrict__ csum) {
  __shared__ float Kt[2][16 * LDK];
  __shared__ float Vt[2][16 * LDK];
  __shared__ float cs[4][64];
  const int bh = blockIdx.x;
  const int b = bh >> 3, h = bh & 7;
  const float* kbase = qkv + (size_t)b * kN * kT + kD + h * kDh;
  const float* vbase = qkv + (size_t)b * kN * kT + 2 * kD + h * kDh;
  const int t = threadIdx.x;
  const int lane = t & 31;
  const int w = t >> 5;
  const int lcol = lane & 15;
  const int koff = (lane < 16) ? 0 : 2;
  const int tile = w * 2;                 // each wave owns 2 of 16 (4x4) tiles
  const int mT = (tile >> 2) * 16;
  const int nT0 = (tile & 3) * 16;
  const int nT1 = ((tile + 1) & 3) * 16;

  const int rs = t >> 4, csg = (t & 15) * 4;  // 16x64 tile: 1 float4 each
  auto stage = [&](int buf, int j0) {
    async_cp16(&Kt[buf][rs * LDK + csg], &kbase[(size_t)(j0 + rs) * kT + csg]);
    async_cp16(&Vt[buf][rs * LDK + csg], &vbase[(size_t)(j0 + rs) * kT + csg]);
  };

  v8f acc0 = {}, acc1 = {};
  float vsum = 0.f;
  const int col = t & 63, q4 = t >> 6;
  stage(0, 0);
  int buf = 0;
  for (int j0 = 0; j0 < kN; j0 += 16) {
    wait_async0();
    __syncthreads();
    if (j0 + 16 < kN) stage(buf ^ 1, j0 + 16);
    float ax[4], ay[4], b0x[4], b0y[4], b1x[4], b1y[4];
#pragma unroll
    for (int jj = 0; jj < 4; ++jj) {
      int j = jj * 4 + koff;
      ax[jj] = Kt[buf][j * LDK + mT + lcol];
      ay[jj] = Kt[buf][(j + 1) * LDK + mT + lcol];
      b0x[jj] = Vt[buf][j * LDK + nT0 + lcol];
      b0y[jj] = Vt[buf][(j + 1) * LDK + nT0 + lcol];
      b1x[jj] = Vt[buf][j * LDK + nT1 + lcol];
      b1y[jj] = Vt[buf][(j + 1) * LDK + nT1 + lcol];
    }
#pragma unroll
    for (int jj = 0; jj < 4; ++jj) {
      v2f a, b0, b1;
      a.x = ax[jj]; a.y = ay[jj];
      b0.x = b0x[jj]; b0.y = b0y[jj];
      b1.x = b1x[jj]; b1.y = b1y[jj];
      acc0 = wmma4(a, b0, acc0);
      acc1 = wmma4(a, b1, acc1);
    }
    // V column-sum partials from the staged tile (rows q4, q4+4, q4+8, q4+12)
#pragma unroll
    for (int ri = 0; ri < 4; ++ri) vsum += Vt[buf][(q4 + ri * 4) * LDK + col];
    __syncthreads();
    buf ^= 1;
  }
  const float scale = 0.125f;  // 64^-0.5
  const int rowOff = (lane < 16) ? 0 : 8;
  float* Sb = S + (size_t)bh * kDh * kDh;
#pragma unroll
  for (int v = 0; v < 8; ++v) {
    Sb[(mT + v + rowOff) * kDh + nT0 + lcol] = acc0[v] * scale;
    Sb[(mT + v + rowOff) * kDh + nT1 + lcol] = acc1[v] * scale;
  }
  cs[q4][col] = vsum;
  __syncthreads();
  if (t < 64) csum[(size_t)bh * kDh + t] = cs[0][t] + cs[1][t] + cs[2][t] + cs[3][t];
}

// ---------------------------------------------------------------------------
// Streaming QK^T row-logsumexp + out = Q@S - L*csum (+residual for stream 1)
// grid: (N/128, B*H, 2); block 256 = 8 waves (16 query rows each)
// ---------------------------------------------------------------------------
__global__ __launch_bounds__(256) void attn_lse(
    const float* __restrict__ qkv, const float* __restrict__ qoir,
    const float* __restrict__ S, const float* __restrict__ csum,
    float* __restrict__ attn_out, float* __restrict__ qnew_out) {
  __shared__ float Qs[128 * LDK];
  __shared__ float Ks[2][16 * LDK];
  __shared__ float Ss[64 * LDK];
  const int qb = blockIdx.x;
  const int bh = blockIdx.y;
  const int strm = blockIdx.z;
  const int b = bh >> 3, h = bh & 7;
  const float* Qb;
  const float* Res;
  float* Ob;
  size_t qStride, oStride;
  if (strm == 0) {
    Qb = qkv + (size_t)b * kN * kT + h * kDh;      qStride = kT;
    Res = nullptr;
    Ob = attn_out + (size_t)b * kN * kD + h * kDh; oStride = kD;
  } else {
    Qb = qoir + (size_t)b * kN * kD + h * kDh;     qStride = kD;
    Res = Qb;                                      // q_new = qout + qoir
    Ob = qnew_out + (size_t)b * kN * kD + h * kDh; oStride = kD;
  }
  const float* Kb = qkv + (size_t)b * kN * kT + kD + h * kDh;

  const int t = threadIdx.x;
  const int lane = t & 31;
  const int w = t >> 5;
  const int lcol = lane & 15;
  const int koff = (lane < 16) ? 0 : 2;
  const int q0 = qb * 128;
  const int rs = t >> 4, csg = (t & 15) * 4;

  // stage 128x64 Q block once (8 float4/thread) + first K tile, all async
#pragma unroll
  for (int i = 0; i < 8; ++i) {
    int rq = rs + i * 16;
    async_cp16(&Qs[rq * LDK + csg], &Qb[(size_t)(q0 + rq) * qStride + csg]);
  }
  async_cp16(&Ks[0][rs * LDK + csg], &Kb[(size_t)rs * kT + csg]);

  const int mW = w * 16;
  const float scale = 0.125f;
  float m[8], s[8];
#pragma unroll
  for (int v = 0; v < 8; ++v) { m[v] = -__builtin_inff(); s[v] = 0.f; }

  int buf = 0;
  for (int kt = 0; kt < kN; kt += 16) {
    wait_async0();
    __syncthreads();
    if (kt + 16 < kN)
      async_cp16(&Ks[buf ^ 1][rs * LDK + csg],
                 &Kb[(size_t)(kt + 16 + rs) * kT + csg]);
    const v2f* Qrow = (const v2f*)&Qs[(mW + lcol) * LDK];
    const v2f* Krow = (const v2f*)&Ks[buf][lcol * LDK];
    v8f acc = {};
#pragma unroll
    for (int half = 0; half < 2; ++half) {
      v2f af[8], bf[8];
#pragma unroll
      for (int kk = 0; kk < 8; ++kk) {
        int kidx = half * 32 + kk * 4 + koff;
        af[kk] = Qrow[kidx >> 1];      // Q fragment (contig d)
        bf[kk] = Krow[kidx >> 1];      // K^T fragment (contig d per key row)
      }
#pragma unroll
      for (int kk = 0; kk < 8; ++kk) acc = wmma4(af[kk], bf[kk], acc);
    }
    // online row max / sum-exp; xor masks 1,2,4,8 reduce within 16-lane halves
#pragma unroll
    for (int v = 0; v < 8; ++v) {
      float val = acc[v] * scale;
      float tm = val;
      tm = fmaxf(tm, __shfl_xor(tm, 1, 32));
      tm = fmaxf(tm, __shfl_xor(tm, 2, 32));
      tm = fmaxf(tm, __shfl_xor(tm, 4, 32));
      tm = fmaxf(tm, __shfl_xor(tm, 8, 32));
      float nm = fmaxf(m[v], tm);
      float p = __expf(val - nm);
      p += __shfl_xor(p, 1, 32);
      p += __shfl_xor(p, 2, 32);
      p += __shfl_xor(p, 4, 32);
      p += __shfl_xor(p, 8, 32);
      s[v] = s[v] * __expf(m[v] - nm) + p;
      m[v] = nm;
    }
    __syncthreads();
    buf ^= 1;
  }
  float L[8];
#pragma unroll
  for (int v = 0; v < 8; ++v) L[v] = m[v] + __logf(s[v]);

  // stage S (64x64) into LDS, then out(16x64) = Q@S - L*csum (+residual)
  const float* Sg = S + (size_t)bh * kDh * kDh;
#pragma unroll
  for (int i = 0; i < 4; ++i) {
    int r = rs + i * 16;
    async_cp16(&Ss[r * LDK + csg], &Sg[(size_t)r * kDh + csg]);
  }
  wait_async0();
  __syncthreads();

  const float* cb = csum + (size_t)bh * kDh;
  const int rowOff = (lane < 16) ? 0 : 8;
#pragma unroll
  for (int nt = 0; nt < 4; ++nt) {
    const int nT = nt * 16;
    const v2f* Qrow = (const v2f*)&Qs[(mW + lcol) * LDK];
    const float* Scol = &Ss[nT + lcol];
    v8f acc = {};
#pragma unroll
    for (int half = 0; half < 2; ++half) {
      v2f af[8];
      float bx[8], by[8];
#pragma unroll
      for (int kk = 0; kk < 8; ++kk) {
        int kidx = half * 32 + kk * 4 + koff;
        af[kk] = Qrow[kidx >> 1];
        bx[kk] = Scol[kidx * LDK];
        by[kk] = Scol[(kidx + 1) * LDK];
      }
#pragma unroll
      for (int kk = 0; kk < 8; ++kk) {
        v2f bb;
        bb.x = bx[kk];
        bb.y = by[kk];
        acc = wmma4(af[kk], bb, acc);
      }
    }
    float cv = cb[nT + lcol];
#pragma unroll
    for (int v = 0; v < 8; ++v) {
      int gi = q0 + mW + v + rowOff;
      float val = acc[v] - L[v] * cv;
      if (Res) val += Res[(size_t)gi * qStride + nT + lcol];
      Ob[(size_t)gi * oStride + nT + lcol] = val;
    }
  }
}

// ---------------------------------------------------------------------------
extern "C" void kernel_launch(void* const* d_in, const int* in_sizes, int n_in,
                              void* d_out, int out_size, void* d_ws,
                              size_t ws_size, hipStream_t stream) {
  (void)in_sizes; (void)n_in; (void)out_size; (void)ws_size;
  const float* x = (const float*)d_in[0];
  const float* qoir = (const float*)d_in[1];
  const float* w_qkv = (const float*)d_in[2];
  const float* w_out = (const float*)d_in[3];

  float* out = (float*)d_out;
  float* x_new = out;                                    // [4,2048,512]
  float* q_new = out + (size_t)kB * kN * kD;             // [4,2048,512]

  float* ws = (float*)d_ws;
  float* qkv = ws;                                       // 4*2048*1536
  float* attn_out = qkv + (size_t)kB * kN * kT;          // 4*2048*512
  float* Sbuf = attn_out + (size_t)kB * kN * kD;         // 32*64*64
  float* csum = Sbuf + (size_t)kB * kH * kDh * kDh;      // 32*64

  // 1) qkv = x @ w_qkv    [8192,512]@[512,1536]
  gemm_f32_wmma<<<dim3(kT / 32, (kB * kN) / 64), 256, 0, stream>>>(
      x, w_qkv, nullptr, qkv, kB * kN, kT, kD);
  // 2) per-head S = scale*K^T V, csum = colsum(V)
  kv_stats<<<dim3(kB * kH), 256, 0, stream>>>(qkv, Sbuf, csum);
  // 3) streaming logsumexp attention; writes attn_out and q_new (w/ residual)
  attn_lse<<<dim3(kN / 128, kB * kH, 2), 256, 0, stream>>>(
      qkv, qoir, Sbuf, csum, attn_out, q_new);
  // 4) x_new = attn_out @ w_out + x
  gemm_f32_wmma<<<dim3(kD / 32, (kB * kN) / 64), 256, 0, stream>>>(
      attn_out, w_out, x, x_new, kB * kN, kD, kD);
}